// Dual_Encoder_45122926412434
// MI455X (gfx1250) — compile-verified
//
#include <hip/hip_runtime.h>

// ---------------------------------------------------------------------------
// Dual TAGConv encoder for MI455X (gfx1250), wave32, f32 WMMA (16x16x4).
// ---------------------------------------------------------------------------

typedef float v2f __attribute__((ext_vector_type(2)));
typedef float v8f __attribute__((ext_vector_type(8)));

#define NNODES 16384
#define FDIM   2000
#define EEDGES 524288
#define HDIM   128
#define LDIM   15
#define KPAIR1 3000   // (3*FDIM)/2
#define KPAIR2 192    // (3*HDIM)/2

__device__ __forceinline__ v8f wmma_f32_k4(v2f a, v2f b, v8f c) {
  // D = A(16x4 f32) * B(4x16 f32) + C(16x16 f32)
  return __builtin_amdgcn_wmma_f32_16x16x4_f32(false, a, false, b,
                                               (short)0, c, false, false);
}

// ---------------------------------------------------------------------------
// Weight packing: Wp[(nt*KP + kp)*16 + n] = { W[2kp][nt*16+n], W[2kp+1][nt*16+n] }
// so each lane's B fragment is a single contiguous float2 load.
// ---------------------------------------------------------------------------
__global__ void pack_w1_kernel(const float* __restrict__ W, v2f* __restrict__ Wp) {
  int idx = blockIdx.x * 256 + threadIdx.x;           // 8 * 3000 * 16 = 384000
  if (idx >= 8 * KPAIR1 * 16) return;
  int n  = idx & 15;
  int kp = (idx >> 4) % KPAIR1;
  int nt = (idx >> 4) / KPAIR1;
  int c  = nt * 16 + n;
  v2f p;
  p.x = W[(size_t)(2 * kp)     * HDIM + c];
  p.y = W[(size_t)(2 * kp + 1) * HDIM + c];
  Wp[idx] = p;
}

__global__ void pack_w2_kernel(const float* __restrict__ W, v2f* __restrict__ Wp) {
  int idx = blockIdx.x * 256 + threadIdx.x;           // 192 * 16 = 3072
  if (idx >= KPAIR2 * 16) return;
  int n  = idx & 15;
  int kp = idx >> 4;
  v2f p;
  p.x = (n < LDIM) ? W[(size_t)(2 * kp)     * LDIM + n] : 0.0f;
  p.y = (n < LDIM) ? W[(size_t)(2 * kp + 1) * LDIM + n] : 0.0f;
  Wp[idx] = p;
}

// ---------------------------------------------------------------------------
// SpMM: y[row[e], :] += w[e] * x[col[e], :]  (COO scatter-add, float4 lanes)
// ---------------------------------------------------------------------------
__global__ void spmm_atomic_kernel(float* __restrict__ y,
                                   const float* __restrict__ x,
                                   const int* __restrict__ row,
                                   const int* __restrict__ col,
                                   const float* __restrict__ w,
                                   int fquads, unsigned long long total) {
  unsigned long long idx =
      (unsigned long long)blockIdx.x * blockDim.x + threadIdx.x;
  if (idx >= total) return;
  int e  = (int)(idx / (unsigned long long)fquads);
  int f4 = (int)(idx % (unsigned long long)fquads);
  int r  = row[e];
  int c  = col[e];
  float ww = w[e];
  int fdim = fquads * 4;
  const float4 xv = ((const float4*)(x + (size_t)c * fdim))[f4];
  float* yp = y + (size_t)r * fdim + 4 * f4;
  atomicAdd(yp + 0, ww * xv.x);
  atomicAdd(yp + 1, ww * xv.y);
  atomicAdd(yp + 2, ww * xv.z);
  atomicAdd(yp + 3, ww * xv.w);
}

// ---------------------------------------------------------------------------
// GEMM1 fused: h = relu( BN( [x|x1|x2] @ W1 + b1 ) )
// M=16384, K=3*2000, N=128.  Wave: 16 rows x 128 cols (8 v8f accumulators).
// ---------------------------------------------------------------------------
__global__ void __launch_bounds__(256)
gemm1_bn_relu_kernel(const float* __restrict__ x,
                     const float* __restrict__ x1,
                     const float* __restrict__ x2,
                     const v2f* __restrict__ Wp,
                     const float* __restrict__ b1,
                     const float* __restrict__ gam,
                     const float* __restrict__ bet,
                     const float* __restrict__ mu,
                     const float* __restrict__ var,
                     float* __restrict__ h) {
  const int lane = threadIdx.x & 31;
  const int wave = threadIdx.x >> 5;
  const int m    = lane & 15;   // A row / B,C col within tile
  const int hi   = lane >> 4;   // K-pair half select
  const int rowBase = blockIdx.x * 128 + wave * 16;
  const int arow    = rowBase + m;

  v8f acc[8];
#pragma unroll
  for (int nt = 0; nt < 8; ++nt) acc[nt] = (v8f){0, 0, 0, 0, 0, 0, 0, 0};

  const float* Xs[3] = {x, x1, x2};
#pragma unroll
  for (int seg = 0; seg < 3; ++seg) {
    const float* X = Xs[seg] + (size_t)arow * FDIM;
    const int kpBase = seg * (FDIM / 2);
    for (int k0 = 0; k0 < FDIM; k0 += 4) {
      v2f a = *(const v2f*)(X + k0 + 2 * hi);
      const int kp = kpBase + (k0 >> 1) + hi;
#pragma unroll
      for (int nt = 0; nt < 8; ++nt) {
        v2f b = Wp[((size_t)nt * KPAIR1 + kp) * 16 + m];
        acc[nt] = wmma_f32_k4(a, b, acc[nt]);
      }
    }
  }

  // Fused bias + batchnorm (moving stats) + relu, then store.
#pragma unroll
  for (int nt = 0; nt < 8; ++nt) {
    const int c = nt * 16 + m;
    const float s  = gam[c] * rsqrtf(var[c] + 1e-3f);
    const float sh = (b1[c] - mu[c]) * s + bet[c];
#pragma unroll
    for (int v = 0; v < 8; ++v) {
      const int r = rowBase + v + 8 * hi;          // C layout: M = vgpr + 8*hi
      const float val = acc[nt][v] * s + sh;
      h[(size_t)r * HDIM + c] = fmaxf(val, 0.0f);
    }
  }
}

// ---------------------------------------------------------------------------
// GEMM2 fused: z = [h|h1|h2] @ W2 + b2 ; out = z or out += z
// M=16384, K=3*128, N=15 (padded to 16, col 15 masked).
// ---------------------------------------------------------------------------
__global__ void __launch_bounds__(256)
gemm2_out_kernel(const float* __restrict__ h,
                 const float* __restrict__ h1,
                 const float* __restrict__ h2,
                 const v2f* __restrict__ Wp,
                 const float* __restrict__ b2,
                 float* __restrict__ out,
                 int accumulate) {
  const int lane = threadIdx.x & 31;
  const int wave = threadIdx.x >> 5;
  const int m    = lane & 15;
  const int hi   = lane >> 4;
  const int rowBase = blockIdx.x * 128 + wave * 16;
  const int arow    = rowBase + m;

  v8f acc = (v8f){0, 0, 0, 0, 0, 0, 0, 0};

  const float* Xs[3] = {h, h1, h2};
#pragma unroll
  for (int seg = 0; seg < 3; ++seg) {
    const float* X = Xs[seg] + (size_t)arow * HDIM;
    const int kpBase = seg * (HDIM / 2);
#pragma unroll
    for (int k0 = 0; k0 < HDIM; k0 += 4) {
      v2f a = *(const v2f*)(X + k0 + 2 * hi);
      v2f b = Wp[(size_t)(kpBase + (k0 >> 1) + hi) * 16 + m];
      acc = wmma_f32_k4(a, b, acc);
    }
  }

  if (m < LDIM) {
    const float bias = b2[m];
#pragma unroll
    for (int v = 0; v < 8; ++v) {
      const int r = rowBase + v + 8 * hi;
      const size_t o = (size_t)r * LDIM + m;
      const float val = acc[v] + bias;
      out[o] = accumulate ? (out[o] + val) : val;
    }
  }
}

// ---------------------------------------------------------------------------
// Host-side per-branch driver (all launches on `stream`; capture-safe).
// ---------------------------------------------------------------------------
static void run_branch(const float* x, const int* erow, const int* ecol,
                       const float* ew,
                       const float* W1, const float* b1, const float* gam,
                       const float* bet, const float* mu, const float* var,
                       const float* W2, const float* b2,
                       float* out, int accumulate,
                       float* x1, float* x2, float* hbuf, float* h1, float* h2,
                       v2f* Wp1, v2f* Wp2, hipStream_t stream) {
  // Pack weights into WMMA-friendly pair-interleaved layout.
  pack_w1_kernel<<<(8 * KPAIR1 * 16 + 255) / 256, 256, 0, stream>>>(W1, Wp1);
  pack_w2_kernel<<<(KPAIR2 * 16 + 255) / 256, 256, 0, stream>>>(W2, Wp2);

  const unsigned long long totF =
      (unsigned long long)EEDGES * (FDIM / 4);
  const unsigned long long totH =
      (unsigned long long)EEDGES * (HDIM / 4);
  const unsigned int blkF = (unsigned int)((totF + 255) / 256);
  const unsigned int blkH = (unsigned int)((totH + 255) / 256);

  // x1 = A x ; x2 = A x1
  hipMemsetAsync(x1, 0, (size_t)NNODES * FDIM * sizeof(float), stream);
  spmm_atomic_kernel<<<blkF, 256, 0, stream>>>(x1, x, erow, ecol, ew,
                                               FDIM / 4, totF);
  hipMemsetAsync(x2, 0, (size_t)NNODES * FDIM * sizeof(float), stream);
  spmm_atomic_kernel<<<blkF, 256, 0, stream>>>(x2, x1, erow, ecol, ew,
                                               FDIM / 4, totF);

  // h = relu(BN([x|x1|x2] @ W1 + b1))
  gemm1_bn_relu_kernel<<<NNODES / 128, 256, 0, stream>>>(
      x, x1, x2, Wp1, b1, gam, bet, mu, var, hbuf);

  // h1 = A h ; h2 = A h1
  hipMemsetAsync(h1, 0, (size_t)NNODES * HDIM * sizeof(float), stream);
  spmm_atomic_kernel<<<blkH, 256, 0, stream>>>(h1, hbuf, erow, ecol, ew,
                                               HDIM / 4, totH);
  hipMemsetAsync(h2, 0, (size_t)NNODES * HDIM * sizeof(float), stream);
  spmm_atomic_kernel<<<blkH, 256, 0, stream>>>(h2, h1, erow, ecol, ew,
                                               HDIM / 4, totH);

  // out (+)= [h|h1|h2] @ W2 + b2
  gemm2_out_kernel<<<NNODES / 128, 256, 0, stream>>>(hbuf, h1, h2, Wp2, b2,
                                                     out, accumulate);
}

extern "C" void kernel_launch(void* const* d_in, const int* in_sizes, int n_in,
                              void* d_out, int out_size, void* d_ws,
                              size_t ws_size, hipStream_t stream) {
  (void)in_sizes; (void)n_in; (void)out_size; (void)ws_size;

  const float* x       = (const float*)d_in[0];
  const int*   dm_row  = (const int*)d_in[1];
  const int*   dm_col  = (const int*)d_in[2];
  const float* dm_w    = (const float*)d_in[3];
  const int*   knn_row = (const int*)d_in[4];
  const int*   knn_col = (const int*)d_in[5];
  const float* knn_w   = (const float*)d_in[6];
  const float* W_dm1   = (const float*)d_in[7];
  const float* b_dm1   = (const float*)d_in[8];
  const float* g_dm    = (const float*)d_in[9];
  const float* be_dm   = (const float*)d_in[10];
  const float* m_dm    = (const float*)d_in[11];
  const float* v_dm    = (const float*)d_in[12];
  const float* W_dm2   = (const float*)d_in[13];
  const float* b_dm2   = (const float*)d_in[14];
  const float* W_knn1  = (const float*)d_in[15];
  const float* b_knn1  = (const float*)d_in[16];
  const float* g_knn   = (const float*)d_in[17];
  const float* be_knn  = (const float*)d_in[18];
  const float* m_knn   = (const float*)d_in[19];
  const float* v_knn   = (const float*)d_in[20];
  const float* W_knn2  = (const float*)d_in[21];
  const float* b_knn2  = (const float*)d_in[22];

  float* out = (float*)d_out;

  // Workspace layout.
  char* ws = (char*)d_ws;
  size_t off = 0;
  float* x1  = (float*)(ws + off); off += (size_t)NNODES * FDIM * 4;
  float* x2  = (float*)(ws + off); off += (size_t)NNODES * FDIM * 4;
  float* hb  = (float*)(ws + off); off += (size_t)NNODES * HDIM * 4;
  float* h1  = (float*)(ws + off); off += (size_t)NNODES * HDIM * 4;
  float* h2  = (float*)(ws + off); off += (size_t)NNODES * HDIM * 4;
  v2f*   Wp1 = (v2f*)(ws + off);   off += (size_t)8 * KPAIR1 * 16 * 8;
  v2f*   Wp2 = (v2f*)(ws + off);   off += (size_t)KPAIR2 * 16 * 8;

  // dm branch writes out, knn branch accumulates.
  run_branch(x, dm_row, dm_col, dm_w, W_dm1, b_dm1, g_dm, be_dm, m_dm, v_dm,
             W_dm2, b_dm2, out, /*accumulate=*/0,
             x1, x2, hb, h1, h2, Wp1, Wp2, stream);
  run_branch(x, knn_row, knn_col, knn_w, W_knn1, b_knn1, g_knn, be_knn, m_knn,
             v_knn, W_knn2, b_knn2, out, /*accumulate=*/1,
             x1, x2, hb, h1, h2, Wp1, Wp2, stream);
}